// LinearFwdDiff_5574867550559
// MI455X (gfx1250) — compile-verified
//
#include <hip/hip_runtime.h>

typedef float v2f __attribute__((ext_vector_type(2)));
typedef float v4f __attribute__((ext_vector_type(4)));
typedef float v8f __attribute__((ext_vector_type(8)));

// Output: out[n][l][o] = weight[o][l], n in [0,128), l,o in [0,1024).
// i.e. 128 broadcast copies of weight^T. Pure write-bandwidth problem
// (512 MB stores, 4 MB reads). WMMA f32 16x16x4 with an identity B operand
// performs the 16x16 register transpose (A-layout in -> D-layout out).
//
// Grid: one block per 16x16 tile of weight^T (64x64 = 4096 blocks).
// Block: 256 threads = 8 waves; each wave writes 16 of the 128 copies.

#define L_DIM 1024
#define N_COPIES 128
#define COPIES_PER_WAVE 16
#define LDS_ROW 20   // 16 + 4 dword pad (keeps 16B alignment, breaks bank aliasing)

__global__ __launch_bounds__(256) void wT_bcast_kernel(
    const float* __restrict__ W,   // [1024][1024] row-major: W[o*1024 + l]
    float* __restrict__ out)       // [128][1024][1024]
{
    __shared__ __align__(16) float tile[16 * LDS_ROW];

    const int lane = threadIdx.x & 31;
    const int waveInBlock = threadIdx.x >> 5;

    const int tileId = blockIdx.x;       // 0..4095
    const int tl = tileId >> 6;          // tile row in W^T (the 'l' dimension)
    const int to = tileId & 63;          // tile col in W^T (the 'o' dimension)

    const int half = lane >> 4;          // 0: lanes 0-15, 1: lanes 16-31
    const int m    = lane & 15;

    // ---- WMMA transpose: D = A * I, A[m,kk] = W^T[tl*16+m, to*16+kk] ----
    // A-layout (16x4 f32, 2 VGPRs): lanes 0-15 hold K=0,1; lanes 16-31 hold K=2,3.
    v8f d = {0.f, 0.f, 0.f, 0.f, 0.f, 0.f, 0.f, 0.f};
#pragma unroll
    for (int c = 0; c < 4; ++c) {
        const int kk0 = 4 * c + 2 * half;                 // this lane's first K
        const float* base = W + (size_t)(to * 16 + kk0) * L_DIM + tl * 16 + m;
        v2f a;
        a.x = base[0];                                     // A[m, kk0]   (coalesced row read)
        a.y = base[L_DIM];                                 // A[m, kk0+1]
        // B = rows 4c..4c+3 of the 16x16 identity, in B-layout
        // (4x16 f32, 2 VGPRs): lane holds column n=m, VGPR v holds k'=2*half+v.
        v2f b;
        b.x = (m == (kk0 + 0)) ? 1.0f : 0.0f;
        b.y = (m == (kk0 + 1)) ? 1.0f : 0.0f;
        d = __builtin_amdgcn_wmma_f32_16x16x4_f32(
                /*neg_a=*/false, a, /*neg_b=*/false, b,
                /*c_mod=*/(short)0, d, /*reuse_a=*/false, /*reuse_b=*/false);
    }
    // d now holds W^T tile in C/D layout:
    // VGPR v, lanes 0-15: row M=v, col N=lane; lanes 16-31: row M=v+8, col N=lane-16.

    // ---- Wave 0 stages the tile into LDS (stride-padded rows) ----
    if (threadIdx.x < 32) {
#pragma unroll
        for (int v = 0; v < 8; ++v) {
            const int mm = v + 8 * half;                  // tile row
            tile[mm * LDS_ROW + m] = d[v];
        }
    }
    __syncthreads();

    // ---- Repack: each lane grabs two aligned float4 (rows r and r+8) ----
    const int r0 = lane >> 2;                             // 0..7
    const int c4 = (lane & 3) * 4;                        // 0,4,8,12
    const v4f p0 = *(const v4f*)&tile[r0 * LDS_ROW + c4];        // rows 0..7
    const v4f p1 = *(const v4f*)&tile[(r0 + 8) * LDS_ROW + c4];  // rows 8..15

    // ---- Stream 16 copies per wave with NT B128 stores ----
    const size_t tileBase = (size_t)(tl * 16) * L_DIM + (size_t)(to * 16);
    float* o0 = out + (size_t)(waveInBlock * COPIES_PER_WAVE) * (L_DIM * L_DIM)
                    + tileBase + (size_t)r0 * L_DIM + c4;
    float* o1 = o0 + (size_t)8 * L_DIM;

#pragma unroll
    for (int n = 0; n < COPIES_PER_WAVE; ++n) {
        __builtin_nontemporal_store(p0, (v4f*)o0);
        __builtin_nontemporal_store(p1, (v4f*)o1);
        o0 += (size_t)L_DIM * L_DIM;
        o1 += (size_t)L_DIM * L_DIM;
    }
}

extern "C" void kernel_launch(void* const* d_in, const int* in_sizes, int n_in,
                              void* d_out, int out_size, void* d_ws, size_t ws_size,
                              hipStream_t stream) {
    (void)in_sizes; (void)n_in; (void)d_ws; (void)ws_size; (void)out_size;
    // d_in[0] = x (unused by the reference's returned value), d_in[1] = weight
    const float* weight = (const float*)d_in[1];
    float* out = (float*)d_out;

    dim3 grid(64 * 64);   // one block per 16x16 tile of weight^T
    dim3 block(256);      // 8 waves
    wT_bcast_kernel<<<grid, block, 0, stream>>>(weight, out);
}